// Attention_82944408420458
// MI455X (gfx1250) — compile-verified
//
#include <hip/hip_runtime.h>

typedef __attribute__((ext_vector_type(2))) float v2f;
typedef __attribute__((ext_vector_type(8))) float v8f;

#define SEQ 2048
#define DIM 64
#define NH  16
#define NB  2
#define NKT (SEQ / 16)   // 128 key tiles of 16
#define WAVES_PER_BLOCK 8

// Per-wave LDS: ptile 16 rows x stride 20 (pad to dodge bank conflicts)
#define PT_STRIDE 20
#define LDS_PER_WAVE (16 * PT_STRIDE)

__global__ __launch_bounds__(256) void attn_fp32_wmma_kernel(
    const float* __restrict__ Q, const float* __restrict__ K,
    const float* __restrict__ V, const float* __restrict__ Adj,
    const unsigned char* __restrict__ Mask,
    float* __restrict__ Out, float* __restrict__ Scores)
{
    __shared__ float lds[WAVES_PER_BLOCK * LDS_PER_WAVE];

    const int lane = threadIdx.x & 31;
    const int wv   = threadIdx.x >> 5;
    float* ptile = &lds[wv * LDS_PER_WAVE];

    const int hi = lane >> 4;    // half-wave: 0 or 1
    const int ln = lane & 15;    // 0..15

    // One wave = one (b, h, 16-row q tile)
    const int tile = blockIdx.x * WAVES_PER_BLOCK + wv;
    const int qt = tile & (NKT - 1);
    const int h  = (tile >> 7) & (NH - 1);
    const int b  = tile >> 11;

    const size_t bh = (size_t)b * NH + h;
    const float* Qp = Q + bh * (size_t)SEQ * DIM + (size_t)qt * 16 * DIM;
    const float* Kp = K + bh * (size_t)SEQ * DIM;
    const float* Vp = V + bh * (size_t)SEQ * DIM;
    const float* Ap = Adj + (size_t)b * SEQ * SEQ + (size_t)qt * 16 * SEQ;
    const unsigned char* Mp = Mask + (size_t)b * SEQ;
    float* Op = Out    + bh * (size_t)SEQ * DIM + (size_t)qt * 16 * DIM;
    float* Sp = Scores + bh * (size_t)SEQ * SEQ + (size_t)qt * 16 * SEQ;

    const float scale = 0.125f;  // 1/sqrt(64)

    // ---- Load Q tile as 16 fp32 A-fragments (K chunks of 4) ----
    // A layout (16x4 fp32): lanes 0-15: M=ln, K = 4c+{0,1}; lanes 16-31: M=ln, K = 4c+{2,3}
    v2f qa[16];
#pragma unroll
    for (int c = 0; c < 16; ++c) {
        qa[c] = *(const v2f*)(Qp + ln * DIM + 4 * c + 2 * hi);
    }

    // ---- Pass 1: logits via WMMA, per-lane online softmax stats ----
    // C/D layout: lane holds column n=ln; VGPR i holds row m = i + 8*hi
    float lmax[8], lsum[8];
#pragma unroll
    for (int i = 0; i < 8; ++i) { lmax[i] = -3.0e38f; lsum[i] = 0.0f; }

    for (int kt = 0; kt < NKT; ++kt) {
        const float* Kt = Kp + (size_t)kt * 16 * DIM;
        v8f acc = {0.f, 0.f, 0.f, 0.f, 0.f, 0.f, 0.f, 0.f};
#pragma unroll
        for (int c = 0; c < 16; ++c) {
            // B layout (4x16 fp32, K rows striped like A): same per-lane address pattern
            v2f bf = *(const v2f*)(Kt + ln * DIM + 4 * c + 2 * hi);
            acc = __builtin_amdgcn_wmma_f32_16x16x4_f32(
                false, qa[c], false, bf, (short)0, acc, false, false);
        }
        const float mvadd = Mp[kt * 16 + ln] ? -1.0e9f : 0.0f;
#pragma unroll
        for (int i = 0; i < 8; ++i) {
            float x  = acc[i] * scale + mvadd;
            float nm = fmaxf(lmax[i], x);
            lsum[i]  = lsum[i] * __expf(lmax[i] - nm) + __expf(x - nm);
            lmax[i]  = nm;
        }
    }

    // ---- Cross-lane butterfly reduce within each 16-lane half (same rows) ----
#pragma unroll
    for (int i = 0; i < 8; ++i) {
        float m = lmax[i], s = lsum[i];
#pragma unroll
        for (int off = 1; off < 16; off <<= 1) {
            float om = __shfl_xor(m, off, 32);
            float os = __shfl_xor(s, off, 32);
            float nm = fmaxf(m, om);
            s = s * __expf(m - nm) + os * __expf(om - nm);
            m = nm;
        }
        lmax[i] = m;
        lsum[i] = 1.0f / s;   // store inverse sum
    }

    // ---- Pass 2: recompute logits, write scores (NT), transpose via LDS, PV ----
    v8f oacc[4];
#pragma unroll
    for (int t = 0; t < 4; ++t) oacc[t] = (v8f){0.f,0.f,0.f,0.f,0.f,0.f,0.f,0.f};

    for (int kt = 0; kt < NKT; ++kt) {
        const float* Kt = Kp + (size_t)kt * 16 * DIM;
        v8f acc = {0.f, 0.f, 0.f, 0.f, 0.f, 0.f, 0.f, 0.f};
#pragma unroll
        for (int c = 0; c < 16; ++c) {
            v2f bf = *(const v2f*)(Kt + ln * DIM + 4 * c + 2 * hi);
            acc = __builtin_amdgcn_wmma_f32_16x16x4_f32(
                false, qa[c], false, bf, (short)0, acc, false, false);
        }
        const float mvadd = Mp[kt * 16 + ln] ? -1.0e9f : 0.0f;

        // scores = softmax + adjoin; stream out non-temporally (536 MB write-once,
        // keep L2 for K/V/adjoin reuse) + stash tile in LDS for the transpose
#pragma unroll
        for (int i = 0; i < 8; ++i) {
            const int m = i + 8 * hi;
            float x  = acc[i] * scale + mvadd;
            float p  = __expf(x - lmax[i]) * lsum[i];
            float sc = p + Ap[m * SEQ + kt * 16 + ln];
            __builtin_nontemporal_store(sc, Sp + (size_t)m * SEQ + kt * 16 + ln);
            ptile[m * PT_STRIDE + ln] = sc;
        }
        __builtin_amdgcn_wave_barrier();   // keep LDS store->load ordering (same wave: DS in-order)

        // PV: P tile (A-fragments from LDS) x V tile (B-fragments from global)
#pragma unroll
        for (int c = 0; c < 4; ++c) {
            v2f pa;
            pa.x = ptile[ln * PT_STRIDE + 4 * c + 2 * hi];
            pa.y = ptile[ln * PT_STRIDE + 4 * c + 2 * hi + 1];
#pragma unroll
            for (int t = 0; t < 4; ++t) {
                // B fragment of V: rows k = kt*16 + 4c + {0,1}+2hi, cols d = 16t + ln
                const float* vb = Vp + (size_t)(kt * 16 + 4 * c + 2 * hi) * DIM + 16 * t + ln;
                v2f bf;
                bf.x = vb[0];
                bf.y = vb[DIM];
                oacc[t] = __builtin_amdgcn_wmma_f32_16x16x4_f32(
                    false, pa, false, bf, (short)0, oacc[t], false, false);
            }
        }
        __builtin_amdgcn_wave_barrier();   // don't let next iter's LDS stores slide over reads
    }

    // ---- Write out tile: element (t,i) -> row m = i+8*hi, col d = 16t+ln ----
#pragma unroll
    for (int t = 0; t < 4; ++t)
#pragma unroll
        for (int i = 0; i < 8; ++i)
            Op[(i + 8 * hi) * DIM + 16 * t + ln] = oacc[t][i];
}

extern "C" void kernel_launch(void* const* d_in, const int* in_sizes, int n_in,
                              void* d_out, int out_size, void* d_ws, size_t ws_size,
                              hipStream_t stream) {
    const float* Q   = (const float*)d_in[0];
    const float* K   = (const float*)d_in[1];
    const float* V   = (const float*)d_in[2];
    const float* Adj = (const float*)d_in[3];
    const unsigned char* Mask = (const unsigned char*)d_in[4];

    float* out_all = (float*)d_out;
    float* Out     = out_all;                                   // [B,H,S,D]
    float* Scores  = out_all + (size_t)NB * NH * SEQ * DIM;     // [B,H,S,S]

    const int total_tiles = NB * NH * (SEQ / 16);               // 4096 waves
    const int blocks = total_tiles / WAVES_PER_BLOCK;           // 512
    attn_fp32_wmma_kernel<<<blocks, 256, 0, stream>>>(Q, K, V, Adj, Mask, Out, Scores);
}